// ROIPool_80547816669750
// MI455X (gfx1250) — compile-verified
//
#include <hip/hip_runtime.h>
#include <hip/hip_bf16.h>
#include <math.h>

// Problem constants (match the reference)
#define PP       7
#define NC       256
#define NH       128
#define NW       128
#define NK       512
#define FSCALE   0.125f
#define NPIX     (NH * NW)        // 16384 pixels per mask / feature plane
#define CHUNK    4096             // pixels staged per round = 16 KB LDS
#define NCHUNK   (NPIX / CHUNK)   // 4 rounds
#define NTHREADS 256              // 8 wave32 per workgroup

__global__ __launch_bounds__(NTHREADS)
void roipool_mask_kernel(const float* __restrict__ feat,   // [B,C,H,W]
                         const float* __restrict__ rois,   // [K,5]
                         const float* __restrict__ masks,  // [K,H,W]
                         float* __restrict__ out)          // [K,C,7,7]
{
  // Double-buffered async staging (2 x 16 KB) + bitmask (2 KB + pad)
  __shared__ __attribute__((aligned(16))) float stage[2][CHUNK];
  __shared__ unsigned int mbits[NPIX / 32 + 2];   // +2 pad: word+1 read at tail

  const int k    = blockIdx.x;
  const int tid  = threadIdx.x;
  const int lane = tid & 31;
  const int wav  = tid >> 5;

  const float* mrow = masks + (size_t)k * NPIX;

  // --- ROI geometry (uniform across block; compiler scalarizes) --------
  const float* r = rois + k * 5;
  const int b  = (int)r[0];
  // jnp.round == round-half-even == rintf under default RTE mode
  const int sw = (int)rintf(r[1] * FSCALE);
  const int sh = (int)rintf(r[2] * FSCALE);
  const int ew = (int)rintf(r[3] * FSCALE);
  const int eh = (int)rintf(r[4] * FSCALE);
  const int roi_w = max(ew - sw + 1, 1);
  const int roi_h = max(eh - sh + 1, 1);

  const float* fbase = feat + (size_t)b * NC * NPIX;

  // Speculative prefetch of each channel's first ROI pixel (global_prefetch_b8).
  if (tid < NC) {
    const float* pf = fbase + (size_t)tid * NPIX
                    + min(max(sh, 0), NH - 1) * NW + min(max(sw, 0), NW - 1);
    __builtin_prefetch(pf, 0, 0);
  }

  // --- Phase 1: double-buffered async DMA of the mask -> LDS bitmask ---
  auto issue_chunk = [&](int ch, int buf) {
#pragma unroll
    for (int j = 0; j < CHUNK / (NTHREADS * 4); ++j) {   // 4 x b128 per thread
      const int v4 = j * NTHREADS + tid;                 // float4 index in chunk
      const float* gsrc = mrow + (size_t)ch * CHUNK + (size_t)v4 * 4;
      // Low 32 bits of a generic LDS pointer == wave-relative LDS byte address.
      unsigned int ldsoff = (unsigned int)(size_t)(&stage[buf][v4 * 4]);
      // CDNA5 async DMA: global -> LDS, tracked by ASYNCcnt.
      asm volatile("global_load_async_to_lds_b128 %0, %1, off"
                   :: "v"(ldsoff), "v"(gsrc) : "memory");
    }
  };

  issue_chunk(0, 0);
  for (int ch = 0; ch < NCHUNK; ++ch) {
    const int cur = ch & 1;
    if (ch + 1 < NCHUNK) {
      issue_chunk(ch + 1, cur ^ 1);     // overlap next DMA with this convert
      // Async loads complete in order: cnt<=4 => the 4 older (current-chunk)
      // per-wave DMAs have landed; the 4 just-issued may still be in flight.
      asm volatile("s_wait_asynccnt 4" ::: "memory");
    } else {
      asm volatile("s_wait_asynccnt 0" ::: "memory");
    }
    __syncthreads();

#pragma unroll
    for (int it = 0; it < CHUNK / NTHREADS; ++it) {      // 16 ballots per chunk
      const float v = stage[cur][it * NTHREADS + tid];
#if __has_builtin(__builtin_amdgcn_ballot_w32)
      const unsigned int bal = __builtin_amdgcn_ballot_w32(v > 0.5f);
#else
      const unsigned int bal = (unsigned int)__ballot(v > 0.5f);
#endif
      if (lane == 0)
        mbits[ch * (CHUNK / 32) + it * (NTHREADS / 32) + wav] = bal;
    }
    __syncthreads();   // reads of stage[cur] done before it is re-targeted
  }

  // --- Phase 2: per-(channel, bin) masked max over the bin rectangle ---
  float* orow = out + (size_t)k * NC * (PP * PP);
  for (int item = tid; item < NC * PP * PP; item += NTHREADS) {
    const int c   = item / (PP * PP);
    const int bin = item - c * (PP * PP);
    const int ph  = bin / PP;
    const int pw  = bin - ph * PP;

    // floor/ceil adaptive bin bounds, clipped to [0, H] / [0, W]
    const int h0 = min(max((ph * roi_h) / PP + sh, 0), NH);
    const int h1 = min(max(((ph + 1) * roi_h + PP - 1) / PP + sh, 0), NH);
    const int w0 = min(max((pw * roi_w) / PP + sw, 0), NW);
    const int w1 = min(max(((pw + 1) * roi_w + PP - 1) / PP + sw, 0), NW);

    const float* fc = fbase + (size_t)c * NPIX;
    float m = -INFINITY;
    for (int h = h0; h < h1; ++h) {
      const float* frow = fc + h * NW;
      // One 64-bit mask window per row (bin width <= 21 bits, bit <= 31,
      // so bit+width <= 52 < 64). Replaces per-pixel LDS reads.
      const int pbase = h * NW + w0;
      const int word  = pbase >> 5;
      const int bit   = pbase & 31;
      const unsigned int wlo = (unsigned int)(
          ((((unsigned long long)mbits[word + 1]) << 32) |
            (unsigned long long)mbits[word]) >> bit);
      int j = 0;
      for (int w = w0; w < w1; ++w, ++j) {
        const float v = frow[w];                       // always in-bounds
        m = fmaxf(m, ((wlo >> j) & 1u) ? v : -INFINITY);
      }
    }
    orow[item] = (m == -INFINITY) ? 0.0f : m;   // empty / fully-masked bin -> 0
  }
}

extern "C" void kernel_launch(void* const* d_in, const int* in_sizes, int n_in,
                              void* d_out, int out_size, void* d_ws, size_t ws_size,
                              hipStream_t stream) {
  (void)in_sizes; (void)n_in; (void)out_size; (void)d_ws; (void)ws_size;
  const float* feat  = (const float*)d_in[0];   // [4,256,128,128] f32
  const float* rois  = (const float*)d_in[1];   // [512,5] f32
  const float* masks = (const float*)d_in[2];   // [512,128,128] f32
  float* out = (float*)d_out;                   // [512,256,7,7] f32

  roipool_mask_kernel<<<dim3(NK), dim3(NTHREADS), 0, stream>>>(feat, rois, masks, out);
}